// MultiResLayer_27178553049146
// MI455X (gfx1250) — compile-verified
//
#include <hip/hip_runtime.h>
#include <stdint.h>

// Problem constants from the reference: B=8, L=8192, D=256, K=2, J=13
#define BSZ   8
#define LSEQ  8192
#define DCH   256
#define JLV   13
#define CH    2              // channels per workgroup (8B contiguous chunk per timestep)
#define NTH   512            // threads per workgroup (16 waves -> 4 waves/SIMD)
#define TROWS (NTH / CH)     // 256 time-rows of threads
#define NT    (LSEQ / TROWS) // 32 timesteps per thread (a,y register resident)

// low 32 bits of a generic pointer to __shared__ = wave-relative LDS byte offset
__device__ __forceinline__ uint32_t lds_off(const void* p) {
    return (uint32_t)(uintptr_t)p;
}

__global__ __launch_bounds__(NTH, 1)
void multires_fused_kernel(const float* __restrict__ x,
                           const float* __restrict__ h0,
                           const float* __restrict__ h1,
                           const float* __restrict__ w,
                           float* __restrict__ out)
{
    __shared__ float buf[2][LSEQ * CH];   // 2 x 64 KB double buffer for 'a'

    const int tid  = threadIdx.x;
    const int cl   = tid & (CH - 1);
    const int trow = tid >> 1;
    const int b    = blockIdx.x / (DCH / CH);
    const int cg   = blockIdx.x % (DCH / CH);
    const int c0   = cg * CH;
    const int c    = c0 + cl;

    // ---- async-stage x[b, :, c0:c0+2] into LDS buffer 0 (8B per timestep) ----
    const float* xbase = x + ((size_t)b * LSEQ) * DCH + c0;
    #pragma unroll 1
    for (int k = 0; k < LSEQ / NTH; ++k) {
        const int t = tid + NTH * k;
        uint64_t g  = (uint64_t)(uintptr_t)(xbase + (size_t)t * DCH);
        uint32_t l  = lds_off(&buf[0][t * CH]);
        asm volatile("global_load_async_to_lds_b64 %0, %1, off"
                     :: "v"(l), "v"(g) : "memory");
    }

    // ---- per-channel constants (overlap with async loads) ----
    const float h00 = h0[c * 2 + 0], h01 = h0[c * 2 + 1];
    const float h10 = h1[c * 2 + 0], h11 = h1[c * 2 + 1];
    const float w0c   = w[c];
    const float wlast = w[(JLV + 1) * DCH + c];
    float wlv[JLV];
    #pragma unroll
    for (int j = 0; j < JLV; ++j) wlv[j] = w[(j + 1) * DCH + c];

    asm volatile("s_wait_asynccnt 0" ::: "memory");
    __syncthreads();

    // ---- register-resident a (own timesteps) and y ----
    float a[NT], y[NT];
    #pragma unroll
    for (int i = 0; i < NT; ++i)
        a[i] = buf[0][(trow + TROWS * i) * CH + cl];

    // levels: a[t] from registers, a[t-d] from LDS buf[p], a' written to buf[1-p]
    #pragma unroll
    for (int j = 0; j < JLV; ++j) {
        const float* __restrict__ rd = buf[j & 1];
        float*       __restrict__ wr = buf[(j & 1) ^ 1];
        const int   d   = 1 << j;
        const float ct  = wlv[j] * h11 + (j == 0 ? w0c : 0.0f); // y-coef of a[t]
        const float ctd = wlv[j] * h10;                          // y-coef of a[t-d]
        #pragma unroll
        for (int i = 0; i < NT; ++i) {
            const int t  = trow + TROWS * i;
            const int ts = t - d;
            const float at   = a[i];
            const float atd  = (ts >= 0) ? rd[ts * CH + cl] : 0.0f;
            const float yv   = fmaf(ct, at, fmaf(ctd, atd, (j == 0) ? 0.0f : y[i]));
            const float anew = fmaf(h01, at, h00 * atd); // h0[c,1]*a[t] + h0[c,0]*a[t-d]
            if (j == JLV - 1) {
                wr[t * CH + cl] = fmaf(wlast, anew, yv); // final y straight to LDS
            } else {
                y[i] = yv;
                a[i] = anew;
                wr[t * CH + cl] = anew;                  // for other threads' a[t-d]
            }
        }
        __syncthreads();   // writes to buf[1-p] done before next level reads them
    }

    // ---- async-store y from LDS buffer 1 (JLV odd -> final writes went to buf[1]) ----
    float* obase = out + ((size_t)b * LSEQ) * DCH + c0;
    #pragma unroll 1
    for (int k = 0; k < LSEQ / NTH; ++k) {
        const int t = tid + NTH * k;
        uint64_t g  = (uint64_t)(uintptr_t)(obase + (size_t)t * DCH);
        uint32_t l  = lds_off(&buf[1][t * CH]);
        asm volatile("global_store_async_from_lds_b64 %0, %1, off"
                     :: "v"(g), "v"(l) : "memory");
    }
    asm volatile("s_wait_asynccnt 0" ::: "memory");
}

extern "C" void kernel_launch(void* const* d_in, const int* in_sizes, int n_in,
                              void* d_out, int out_size, void* d_ws, size_t ws_size,
                              hipStream_t stream) {
    (void)in_sizes; (void)n_in; (void)out_size; (void)d_ws; (void)ws_size;
    const float* x  = (const float*)d_in[0];
    const float* h0 = (const float*)d_in[1];
    const float* h1 = (const float*)d_in[2];
    const float* w  = (const float*)d_in[3];
    float* out = (float*)d_out;

    dim3 grid(BSZ * (DCH / CH));   // 8 * 128 = 1024 workgroups
    multires_fused_kernel<<<grid, NTH, 0, stream>>>(x, h0, h1, w, out);
}